// PointsSamplerDensityAndManhattanMeanwhile_54898271977790
// MI455X (gfx1250) — compile-verified
//
#include <hip/hip_runtime.h>
#include <hip/hip_bf16.h>

#define BATCH   4
#define NPTS    8192
#define NPOINT  2048
#define RADIUS  0.5f
#define TILE    16
#define NTILES  (NPTS / TILE)     // 512
#define THREADS 1024
#define PPT     (NPTS / THREADS)  // 8 points per thread

typedef __attribute__((ext_vector_type(2))) float v2f;
typedef __attribute__((ext_vector_type(8))) float v8f;

// -----------------------------------------------------------------------------
// Kernel 1: ball-count density via V_WMMA_F32_16X16X4_F32.
//   A row m (database pt) = ( x, y, z, |p|^2 )
//   B col n (query pt)    = (-2x, -2y, -2z, 1 )
//   D[m][n] = |p_m|^2 - 2 p_m.q_n
//   count condition:  D[m][n] < r^2 - |q_n|^2   (per-lane hoisted threshold)
// One wave per 16-query column tile; loops over all 512 row tiles.
// Writes dweight = density^(-1) (alpha = 1) to workspace.
// -----------------------------------------------------------------------------
__global__ __launch_bounds__(256)
void density_wmma_kernel(const float* __restrict__ pts, float* __restrict__ dweight)
{
    const int lane = threadIdx.x & 31;
    const int wid  = (blockIdx.x * blockDim.x + threadIdx.x) >> 5;  // global wave id
    const int b    = wid / NTILES;
    const int ct   = wid % NTILES;          // column tile of 16 query points
    const float* __restrict__ P = pts + (size_t)b * NPTS * 3;

    const int  m  = lane & 15;
    const bool hi = (lane >= 16);

    // Per-lane query point (column n = lane % 16)
    const int   cp  = ct * TILE + m;
    const float qx  = P[cp * 3 + 0];
    const float qy  = P[cp * 3 + 1];
    const float qz  = P[cp * 3 + 2];
    const float qn2 = qx * qx + qy * qy + qz * qz;

    // B matrix 4x16 (2 VGPRs): lanes<16 hold K=0,1 ; lanes>=16 hold K=2,3
    v2f bm;
    bm.x = hi ? (-2.0f * qz) : (-2.0f * qx);
    bm.y = hi ? 1.0f         : (-2.0f * qy);

    // Hoisted per-lane threshold: D < r^2 - |q|^2
    const float thr = RADIUS * RADIUS - qn2;
    unsigned    cnt = 0u;

    for (int rt = 0; rt < NTILES; ++rt) {
        const int   rp = rt * TILE + m;
        const float ax = P[rp * 3 + 0];
        const float ay = P[rp * 3 + 1];
        const float az = P[rp * 3 + 2];
        // Speculative prefetch of the next tile (safe past the end on gfx1250).
        __builtin_prefetch(&P[(rp + TILE) * 3], 0, 1);       // global_prefetch_b8
        const float an2 = ax * ax + ay * ay + az * az;

        // A matrix 16x4 (2 VGPRs): lanes<16 hold K=0,1 ; lanes>=16 hold K=2,3
        v2f am;
        am.x = hi ? az  : ax;
        am.y = hi ? an2 : ay;

        v8f c = {};
        v8f dmat = __builtin_amdgcn_wmma_f32_16x16x4_f32(
            /*neg_a=*/false, am, /*neg_b=*/false, bm,
            /*c_mod=*/(short)0, c, /*reuse_a=*/false, /*reuse_b=*/false);

        // Lane holds column n for 8 rows; count D < thr
        #pragma unroll
        for (int v = 0; v < 8; ++v)
            cnt += (dmat[v] < thr) ? 1u : 0u;
    }

    // Lane L and L^16 hold rows 0-7 / 8-15 for the same column: combine halves.
    cnt += (unsigned)__shfl_xor((int)cnt, 16);
    if (!hi)
        dweight[(size_t)b * NPTS + cp] = 1.0f / (float)cnt;  // density^(-alpha), alpha=1
}

// -----------------------------------------------------------------------------
// Kernel 2: density-weighted Manhattan FPS. One workgroup per batch.
// 1024 threads x 8 register-resident points; 2047 sequential iterations with
// wave32 shuffle argmax + 32-entry LDS cross-wave reduce (2 barriers/iter).
// Tie-break: lowest index wins (matches jnp.argmax).
// -----------------------------------------------------------------------------
__global__ __launch_bounds__(THREADS)
void fps_kernel(const float* __restrict__ pts,
                const float* __restrict__ dweight,
                int* __restrict__ out)
{
    __shared__ float s_rv[32];
    __shared__ int   s_ri[32];
    __shared__ int   s_win;

    const int b    = blockIdx.x;
    const int t    = threadIdx.x;
    const int lane = t & 31;
    const int wv   = t >> 5;
    const float* __restrict__ P  = pts     + (size_t)b * NPTS * 3;
    const float* __restrict__ DW = dweight + (size_t)b * NPTS;

    float px[PPT], py[PPT], pz[PPT], md[PPT], dw[PPT];
    #pragma unroll
    for (int k = 0; k < PPT; ++k) {
        const int p = k * THREADS + t;          // strided ownership, coalesced
        px[k] = P[p * 3 + 0];
        py[k] = P[p * 3 + 1];
        pz[k] = P[p * 3 + 2];
        dw[k] = DW[p];
        md[k] = 1e10f;
    }

    if (t == 0) out[b * NPOINT] = 0;            // first pick = point 0

    int w = 0;                                  // last selected index
    for (int it = 1; it < NPOINT; ++it) {
        // Uniform-address broadcast load of last pick's coords (L2-hot).
        const float lx = P[w * 3 + 0];
        const float ly = P[w * 3 + 1];
        const float lz = P[w * 3 + 2];

        float bv = -1.0f;
        int   bi = 0x7fffffff;
        #pragma unroll
        for (int k = 0; k < PPT; ++k) {
            const float d  = fabsf(px[k] - lx) + fabsf(py[k] - ly) + fabsf(pz[k] - lz);
            md[k]          = fminf(md[k], d);
            const float sc = md[k] * dw[k];
            const int   p  = k * THREADS + t;
            if (sc > bv || (sc == bv && p < bi)) { bv = sc; bi = p; }
        }

        // wave32 argmax (prefer lower index on ties)
        #pragma unroll
        for (int off = 16; off > 0; off >>= 1) {
            const float ov = __shfl_xor(bv, off);
            const int   oi = __shfl_xor(bi, off);
            if (ov > bv || (ov == bv && oi < bi)) { bv = ov; bi = oi; }
        }
        if (lane == 0) { s_rv[wv] = bv; s_ri[wv] = bi; }
        __syncthreads();

        if (t < 32) {
            float v2 = s_rv[lane];
            int   i2 = s_ri[lane];
            #pragma unroll
            for (int off = 16; off > 0; off >>= 1) {
                const float ov = __shfl_xor(v2, off);
                const int   oi = __shfl_xor(i2, off);
                if (ov > v2 || (ov == v2 && oi < i2)) { v2 = ov; i2 = oi; }
            }
            if (t == 0) { s_win = i2; out[b * NPOINT + it] = i2; }
        }
        __syncthreads();
        w = s_win;
    }
}

// -----------------------------------------------------------------------------
extern "C" void kernel_launch(void* const* d_in, const int* in_sizes, int n_in,
                              void* d_out, int out_size, void* d_ws, size_t ws_size,
                              hipStream_t stream)
{
    const float* pts = (const float*)d_in[0];   // [4, 8192, 3] f32
    // d_in[1] (features) is unused by D-FPS, matching the reference.
    float* dw  = (float*)d_ws;                  // [4, 8192] dweight scratch (128 KB)
    int*   out = (int*)d_out;                   // [4, 2048] int32 indices

    const int total_waves = BATCH * NTILES;                 // 2048 waves
    const int blocks      = (total_waves * 32) / 256;       // 256 blocks x 256 thr
    density_wmma_kernel<<<blocks, 256, 0, stream>>>(pts, dw);
    fps_kernel<<<BATCH, THREADS, 0, stream>>>(pts, dw, out);
}